// KalmanFilterRNN_10943576670396
// MI455X (gfx1250) — compile-verified
//
#include <hip/hip_runtime.h>
#include <cstdint>
#include <cstddef>

// ---------------- problem constants (from the reference) ----------------
// A = I + DT*A0, DT=0.1, B/M=0.1, K/M=4.0:
//   A = [[a,0,b,0],[0,a,0,b],[c,0,1,0],[0,c,0,1]],  a=0.99, b=-0.4, c=0.1
constexpr int   kT = 1000;
constexpr int   kN = 8192;
constexpr float A_a = 0.99f;
constexpr float A_b = -0.4f;
constexpr float A_c = 0.1f;

constexpr int BLOCK  = 32;   // one wave32 per workgroup -> spread over max WGPs
constexpr int STAGES = 8;    // async global->LDS pipeline depth
constexpr int OPS_PER_STAGE = 3;  // z:b64 + L:b64 + L:b32(offset:8)

// bytes per timestep row
constexpr size_t Z_TSTRIDE = (size_t)kN * 2 * sizeof(float);
constexpr size_t L_TSTRIDE = (size_t)kN * 3 * sizeof(float);
// LDS bytes per stage
constexpr uint32_t ZL_SSTRIDE = BLOCK * 2 * sizeof(float);
constexpr uint32_t LL_SSTRIDE = BLOCK * 3 * sizeof(float);

// ---------------- CDNA5 async-copy primitives (gfx1250) ----------------
__device__ __forceinline__ uint32_t lds_off(const void* p) {
    // generic LDS pointer = {shared_aperture_hi, lds_offset_lo}; low 32 bits
    // are exactly the wave-relative LDS address the async engine wants.
    return (uint32_t)(uintptr_t)p;
}

template <int N>
__device__ __forceinline__ void wait_asynccnt_le() {
    asm volatile("s_wait_asynccnt %0" :: "i"(N) : "memory");
}
__device__ __forceinline__ void wait_dscnt0() {
    asm volatile("s_wait_dscnt 0" ::: "memory");
}

// One pipeline stage: z (8B/lane) + L_hat (12B/lane), all non-temporal.
// NOTE: async-load IOFFSET is added to BOTH the LDS and global addresses,
// so the L tail reuses the same two address registers with offset:8.
__device__ __forceinline__ void issue_stage(uint32_t zl, uint32_t ll,
                                            const char* zg, const char* lg) {
    asm volatile("global_load_async_to_lds_b64 %0, %1, off th:TH_LOAD_NT"
                 :: "v"(zl), "v"(zg) : "memory");
    asm volatile("global_load_async_to_lds_b64 %0, %1, off th:TH_LOAD_NT"
                 :: "v"(ll), "v"(lg) : "memory");
    asm volatile("global_load_async_to_lds_b32 %0, %1, off offset:8 th:TH_LOAD_NT"
                 :: "v"(ll), "v"(lg) : "memory");
}

// ---------------- kernel ----------------
__global__ __launch_bounds__(BLOCK)
void KalmanFilterRNN_10943576670396_kernel(const float* __restrict__ z,
                                           const float* __restrict__ Lh,
                                           float* __restrict__ ys) {
    __shared__ __align__(16) float zbuf[STAGES][BLOCK * 2];
    __shared__ __align__(16) float lbuf[STAGES][BLOCK * 3];

    const int tid = threadIdx.x;
    const int n   = blockIdx.x * BLOCK + tid;

    const char* zg = (const char*)(z  + (size_t)n * 2);
    const char* lg = (const char*)(Lh + (size_t)n * 3);
    const uint32_t zl = lds_off(&zbuf[0][tid * 2]);
    const uint32_t ll = lds_off(&lbuf[0][tid * 3]);

    // ---- prologue: fill the whole pipeline (24 async ops in flight) ----
#pragma unroll
    for (int s = 0; s < STAGES; ++s) {
        issue_stage(zl + (uint32_t)s * ZL_SSTRIDE,
                    ll + (uint32_t)s * LL_SSTRIDE,
                    zg + (size_t)s * Z_TSTRIDE,
                    lg + (size_t)s * L_TSTRIDE);
    }
    const char* zg_next = zg + (size_t)STAGES * Z_TSTRIDE;
    const char* lg_next = lg + (size_t)STAGES * L_TSTRIDE;

    // ---- filter state in registers ----
    float mu0 = 0.f, mu1 = 0.f, mu2 = 1.f, mu3 = 1.f;
    // symmetric Sigma (init = I)
    float p00 = 1.f, p01 = 0.f, p02 = 0.f, p03 = 0.f;
    float p11 = 1.f, p12 = 0.f, p13 = 0.f;
    float p22 = 1.f, p23 = 0.f, p33 = 1.f;

    // running non-temporal output pointer: ys[t][n][0:2] as one b64 store
    unsigned long long* yp = (unsigned long long*)ys + n;

    // shared per-step body (predict + update + emit)
    auto body = [&](float z0, float z1, float L0, float L1, float L2) {
        // ---- predict mean: mu_pred = A mu ----
        const float m0 = A_a * mu0 + A_b * mu2;
        const float m1 = A_a * mu1 + A_b * mu3;
        const float m2 = A_c * mu0 + mu2;
        const float m3 = A_c * mu1 + mu3;

        // ---- predict covariance: Sp = A P A^T + diag(1,1,0,0) ----
        const float u0 = A_a * p00 + A_b * p02, u2 = A_a * p02 + A_b * p22;
        const float v0 = A_a * p01 + A_b * p03, v1 = A_a * p11 + A_b * p13;
        const float v2 = A_a * p12 + A_b * p23, v3 = A_a * p13 + A_b * p33;
        const float w0 = A_c * p00 + p02, w1 = A_c * p01 + p12;
        const float w2 = A_c * p02 + p22, w3 = A_c * p03 + p23;
        const float x0 = A_c * p01 + p03, x1 = A_c * p11 + p13;
        const float x2 = A_c * p12 + p23, x3 = A_c * p13 + p33;

        const float s00 = A_a * u0 + A_b * u2 + 1.0f;
        const float s01 = A_a * v0 + A_b * v2;
        const float s02 = A_a * w0 + A_b * w2;
        const float s03 = A_a * x0 + A_b * x2;
        const float s11 = A_a * v1 + A_b * v3 + 1.0f;
        const float s12 = A_a * w1 + A_b * w3;
        const float s13 = A_a * x1 + A_b * x3;
        const float s22 = A_c * w0 + w2;
        const float s23 = A_c * x0 + x2;
        const float s33 = A_c * x1 + x3;

        // ---- measurement covariance R = L L^T ----
        const float l00 = __expf(L0);
        const float l10 = L1;
        const float l11 = __expf(L2);
        const float r00 = l00 * l00;
        const float r01 = l00 * l10;
        const float r11 = l10 * l10 + l11 * l11;

        // ---- innovation covariance + 2x2 inverse ----
        const float S00 = s22 + r00;
        const float S01 = s23 + r01;
        const float S11 = s33 + r11;
        const float det = S00 * S11 - S01 * S01;
        float inv = __builtin_amdgcn_rcpf(det);
        inv = inv * (2.0f - det * inv);  // one Newton step
        const float si00 =  S11 * inv;
        const float si01 = -S01 * inv;
        const float si11 =  S00 * inv;

        // ---- Kalman gain K = Sp[:,2:4] @ Sinv ----
        const float k00 = s02 * si00 + s03 * si01;
        const float k01 = s02 * si01 + s03 * si11;
        const float k10 = s12 * si00 + s13 * si01;
        const float k11 = s12 * si01 + s13 * si11;
        const float k20 = s22 * si00 + s23 * si01;
        const float k21 = s22 * si01 + s23 * si11;
        const float k30 = s23 * si00 + s33 * si01;
        const float k31 = s23 * si01 + s33 * si11;

        // ---- state update ----
        const float i0 = z0 - m2;
        const float i1 = z1 - m3;
        mu0 = m0 + k00 * i0 + k01 * i1;
        mu1 = m1 + k10 * i0 + k11 * i1;
        mu2 = m2 + k20 * i0 + k21 * i1;
        mu3 = m3 + k30 * i0 + k31 * i1;

        // ---- covariance update: P = Sp - K @ Sp[2:4,:] (symmetric) ----
        p00 = s00 - k00 * s02 - k01 * s03;
        p01 = s01 - k00 * s12 - k01 * s13;
        p02 = s02 - k00 * s22 - k01 * s23;
        p03 = s03 - k00 * s23 - k01 * s33;
        p11 = s11 - k10 * s12 - k11 * s13;
        p12 = s12 - k10 * s22 - k11 * s23;
        p13 = s13 - k10 * s23 - k11 * s33;
        p22 = s22 - k20 * s22 - k21 * s23;
        p23 = s23 - k20 * s23 - k21 * s33;
        p33 = s33 - k30 * s23 - k31 * s33;

        // ---- emit y = mu[2:4] as one non-temporal b64 store ----
        union { float f[2]; unsigned long long u; } pk;
        pk.f[0] = mu2;
        pk.f[1] = mu3;
        __builtin_nontemporal_store(pk.u, yp);
        yp += kN;
    };

    // ---- main loop: always 24 async ops in flight, branch-free issue ----
    for (int t = 0; t < kT - STAGES; ++t) {
        const int s = t & (STAGES - 1);
        const uint32_t zls = zl + (uint32_t)s * ZL_SSTRIDE;
        const uint32_t lls = ll + (uint32_t)s * LL_SSTRIDE;

        // oldest stage's 3 async ops complete (in-order per wave)
        wait_asynccnt_le<OPS_PER_STAGE * (STAGES - 1)>();

        const float2 zt = *(const float2*)&zbuf[s][tid * 2];
        const float  L0 = lbuf[s][tid * 3 + 0];
        const float  L1 = lbuf[s][tid * 3 + 1];
        const float  L2 = lbuf[s][tid * 3 + 2];
        wait_dscnt0();  // slot fully read before the refill overwrites it

        issue_stage(zls, lls, zg_next, lg_next);
        zg_next += Z_TSTRIDE;
        lg_next += L_TSTRIDE;

        body(zt.x, zt.y, L0, L1, L2);
    }

    // ---- tail: drain the pipeline; wait for ALL outstanding async ops so
    // every remaining stage is guaranteed resident before its read ----
    wait_asynccnt_le<0>();
    for (int t = kT - STAGES; t < kT; ++t) {
        const int s = t & (STAGES - 1);
        const float2 zt = *(const float2*)&zbuf[s][tid * 2];
        const float  L0 = lbuf[s][tid * 3 + 0];
        const float  L1 = lbuf[s][tid * 3 + 1];
        const float  L2 = lbuf[s][tid * 3 + 2];
        body(zt.x, zt.y, L0, L1, L2);
    }
}

// ---------------- launch ----------------
extern "C" void kernel_launch(void* const* d_in, const int* in_sizes, int n_in,
                              void* d_out, int out_size, void* d_ws, size_t ws_size,
                              hipStream_t stream) {
    (void)in_sizes; (void)n_in; (void)out_size; (void)d_ws; (void)ws_size;
    const float* z  = (const float*)d_in[0];   // (T, N, 2) f32
    const float* Lh = (const float*)d_in[1];   // (T, N, 3) f32
    float* ys = (float*)d_out;                 // (T, N, 2) f32

    dim3 grid(kN / BLOCK), block(BLOCK);
    hipLaunchKernelGGL(KalmanFilterRNN_10943576670396_kernel,
                       grid, block, 0, stream, z, Lh, ys);
}